// UnifiedCTNN_18056042512512
// MI455X (gfx1250) — compile-verified
//
#include <hip/hip_runtime.h>
#include <hip/hip_bf16.h>

typedef __attribute__((ext_vector_type(2))) float v2f;
typedef __attribute__((ext_vector_type(8))) float v8f;

#define SQRT_OMEGA 0.7071067811865476f

__device__ __forceinline__ float silu_f(float v) {
  return v / (1.0f + __expf(-v));
}

__device__ __forceinline__ v8f wmma4(v2f a, v2f b, v8f c) {
  // D(16x16,f32) += A(16x4,f32) * B(4x16,f32)
  return __builtin_amdgcn_wmma_f32_16x16x4_f32(false, a, false, b, (short)0, c,
                                               false, false);
}

// One K=128 GEMM segment over FOUR 16-row tiles at once (acc[0..3]).
// K-step is the outer loop so each weight B-fragment is loaded once and reused
// by 4 WMMAs; consecutive WMMAs hit different accumulators (D->C chain spaced 4).
// W stored (out_dim, in_dim) row-major => B[k][n] = W[col0+n][kOffW+k].
__device__ __forceinline__ void seg128(v8f acc[4], const float* A, int lda,
                                       const float* W, int ldw, int col0,
                                       int kOffW) {
  const int lane = threadIdx.x & 31;
  const int l15 = lane & 15;
  const int half = lane >> 4;
  const float* wrow = W + (size_t)(col0 + l15) * ldw + kOffW + 2 * half;
  const float* abase = A + (size_t)l15 * lda + 2 * half;
#pragma unroll 8
  for (int k = 0; k < 128; k += 4) {
    v2f w; w[0] = wrow[k]; w[1] = wrow[k + 1];
#pragma unroll
    for (int rt = 0; rt < 4; ++rt) {
      const float* arow = abase + (size_t)(rt * 16) * lda + k;
      v2f a; a[0] = arow[0]; a[1] = arow[1];
      acc[rt] = wmma4(a, w, acc[rt]);
    }
  }
}

// Same, but A is one 128-vector broadcast to all rows (v_i segment): both
// fragments are row-tile-invariant -> loaded once per K-step, 4 WMMAs each.
__device__ __forceinline__ void seg128_bcast(v8f acc[4], const float* avec,
                                             const float* W, int ldw, int col0,
                                             int kOffW) {
  const int lane = threadIdx.x & 31;
  const int l15 = lane & 15;
  const int half = lane >> 4;
  const float* wrow = W + (size_t)(col0 + l15) * ldw + kOffW + 2 * half;
  const float* arow = avec + 2 * half;
#pragma unroll 8
  for (int k = 0; k < 128; k += 4) {
    v2f w; w[0] = wrow[k]; w[1] = wrow[k + 1];
    v2f a; a[0] = arow[k]; a[1] = arow[k + 1];
#pragma unroll
    for (int rt = 0; rt < 4; ++rt) {
      acc[rt] = wmma4(a, w, acc[rt]);
    }
  }
}

// C/D layout: VGPR r holds (row0+r, col) on lanes 0-15 and (row0+8+r, col) on 16-31.
__device__ __forceinline__ void store_tile(float* dst, int ldd, int row0, int col0,
                                           v8f acc, const float* bias,
                                           const float* resid, int ldr, int act) {
  const int lane = threadIdx.x & 31;
  const int l15 = lane & 15;
  const int half = lane >> 4;
  const int col = col0 + l15;
  const float bv = bias ? bias[col] : 0.0f;
#pragma unroll
  for (int r = 0; r < 8; ++r) {
    const int row = row0 + half * 8 + r;
    float vv = acc[r] + bv;
    if (resid) vv += resid[(size_t)row * ldr + col];
    if (act) vv = silu_f(vv);
    dst[(size_t)row * ldd + col] = vv;
  }
}

__device__ __forceinline__ void zero4(v8f acc[4]) {
#pragma unroll
  for (int rt = 0; rt < 4; ++rt) acc[rt] = (v8f){};
}

// ---------------- Kernel 1: node embedding + v = rho_v_to_e(h_v) --------------
__global__ __launch_bounds__(256) void k1_node(const float* __restrict__ x,
                                               const int* __restrict__ spin,
                                               const float* ne_w, const float* ne_b,
                                               const float* rv_w,
                                               float* hv, float* v) {
  __shared__ float s_hv[64 * 128];
  const int b = blockIdx.x;
  const int tid = threadIdx.x;
  for (int idx = tid; idx < 64 * 128; idx += 256) {
    const int row = idx >> 7;  // node i
    const int c = idx & 127;
    const float xs0 = x[(b * 64 + row) * 2 + 0] * SQRT_OMEGA;
    const float xs1 = x[(b * 64 + row) * 2 + 1] * SQRT_OMEGA;
    const float sp = (float)spin[row];
    const float vv = ne_b[c] + ne_w[c * 3 + 0] * xs0 + ne_w[c * 3 + 1] * xs1 +
                     ne_w[c * 3 + 2] * sp;
    s_hv[idx] = vv;
    hv[(size_t)b * 8192 + idx] = vv;
  }
  __syncthreads();
  const int wave = tid >> 5;
  const int col0 = wave * 16;
  v8f acc[4];
  zero4(acc);
  seg128(acc, s_hv, 128, rv_w, 128, col0, 0);
  for (int rt = 0; rt < 4; ++rt)
    store_tile(v + (size_t)b * 8192, 128, rt * 16, col0, acc[rt], nullptr,
               nullptr, 0, 0);
}

// -------- Kernel 2: fused edge pipeline + masked j-aggregation (the hot loop) --
// One block per (b,i): 64 edge rows j, h_e never touches global memory.
__global__ __launch_bounds__(256) void k2_edge(const float* __restrict__ x,
                                               const float* __restrict__ v,
                                               const float* ee_w0, const float* ee_b0,
                                               const float* ee_w1, const float* ee_b1,
                                               const float* eu_w0, const float* eu_b0,
                                               const float* eu_w1, const float* eu_b1,
                                               float* smv, float* stri) {
  __shared__ float s_he0[64 * 128];
  __shared__ float s_t[64 * 128];
  const int b = blockIdx.x >> 6;
  const int i = blockIdx.x & 63;
  const int tid = threadIdx.x;
  const int wave = tid >> 5;
  const int lane = tid & 31;
  const int l15 = lane & 15;
  const int half = lane >> 4;
  const int col0 = wave * 16;

  // Warm caches for the big edge-update weight matrix (global_prefetch_b8).
  __builtin_prefetch(eu_w0 + (size_t)tid * 192, 0, 3);

  const float xi0 = x[(b * 64 + i) * 2 + 0] * SQRT_OMEGA;
  const float xi1 = x[(b * 64 + i) * 2 + 1] * SQRT_OMEGA;

  v8f acc[4];

  // ---- edge embed layer 0: edge_in(4) -> 128, silu, into s_t ----
  {
    const float* wrow = ee_w0 + (col0 + l15) * 4 + 2 * half;
    v2f bw; bw[0] = wrow[0]; bw[1] = wrow[1];
    zero4(acc);
#pragma unroll
    for (int rt = 0; rt < 4; ++rt) {
      const int j = rt * 16 + l15;
      const float xj0 = x[(b * 64 + j) * 2 + 0] * SQRT_OMEGA;
      const float xj1 = x[(b * 64 + j) * 2 + 1] * SQRT_OMEGA;
      const float rv0 = xi0 - xj0;
      const float rv1 = xi1 - xj1;
      v2f a;
      if (half == 0) {  // features 0,1 = r_vec
        a[0] = rv0; a[1] = rv1;
      } else {          // features 2,3 = r1, r2
        const float r2 = rv0 * rv0 + rv1 * rv1;
        a[0] = sqrtf(r2 + 1e-12f);
        a[1] = r2;
      }
      acc[rt] = wmma4(a, bw, acc[rt]);
    }
    for (int rt = 0; rt < 4; ++rt)
      store_tile(s_t, 128, rt * 16, col0, acc[rt], ee_b0, nullptr, 0, 1);
  }
  __syncthreads();
  // ---- edge embed layer 1: 128 -> 128, into s_he0 ----
  zero4(acc);
  seg128(acc, s_t, 128, ee_w1, 128, col0, 0);
  for (int rt = 0; rt < 4; ++rt)
    store_tile(s_he0, 128, rt * 16, col0, acc[rt], ee_b1, nullptr, 0, 0);
  __syncthreads();
  // ---- edge update layer 0: [h_e0 | v_i | v_j](384) -> 128, silu, into s_t ----
  const float* v_b = v + (size_t)b * 8192;
  const float* v_i = v_b + i * 128;
  zero4(acc);
  seg128(acc, s_he0, 128, eu_w0, 384, col0, 0);
  seg128_bcast(acc, v_i, eu_w0, 384, col0, 128);
  seg128(acc, v_b, 128, eu_w0, 384, col0, 256);
  for (int rt = 0; rt < 4; ++rt)
    store_tile(s_t, 128, rt * 16, col0, acc[rt], eu_b0, nullptr, 0, 1);
  __syncthreads();
  // ---- edge update layer 1 + residual + masked sums over j (register-only,
  //      deterministic: each wave owns its 16 columns for all 64 rows) ----
  zero4(acc);
  seg128(acc, s_t, 128, eu_w1, 128, col0, 0);
  float pmv = 0.0f, ptri = 0.0f;
  const int col = col0 + l15;
  const float bsv = eu_b1[col];
  for (int rt = 0; rt < 4; ++rt) {
#pragma unroll
    for (int r = 0; r < 8; ++r) {
      const int j = rt * 16 + half * 8 + r;
      const float he = acc[rt][r] + bsv + s_he0[j * 128 + col];
      if (j != i) pmv += he;   // m_v aggregation (off-diagonal)
      if (j > i) ptri += he;   // upper-triangular sum for Jastrow mean
    }
  }
  pmv += __shfl_xor(pmv, 16, 32);
  ptri += __shfl_xor(ptri, 16, 32);
  if (lane < 16) {
    smv[((size_t)b * 64 + i) * 128 + col] = pmv;
    stri[((size_t)b * 64 + i) * 128 + col] = ptri;
  }
}

// ------------- Kernel 3: node update MLP + residual + dx head -----------------
__global__ __launch_bounds__(256) void k3_nodeupd(const float* __restrict__ hv,
                                                  const float* __restrict__ smv,
                                                  const float* re_w,
                                                  const float* nu_w0, const float* nu_b0,
                                                  const float* nu_w1, const float* nu_b1,
                                                  const float* nu_w2, const float* nu_b2,
                                                  const float* dx_w, const float* dx_b,
                                                  float* hvn, float* dxr) {
  __shared__ float sA[64 * 128];
  __shared__ float sB[64 * 128];
  const int b = blockIdx.x;
  const int tid = threadIdx.x;
  const int wave = tid >> 5;
  const int col0 = wave * 16;
  const float* hv_b = hv + (size_t)b * 8192;
  v8f acc[4];
  // m_v = re_w @ (sum_j h_e)  -> sA
  zero4(acc);
  seg128(acc, smv + (size_t)b * 8192, 128, re_w, 128, col0, 0);
  for (int rt = 0; rt < 4; ++rt)
    store_tile(sA, 128, rt * 16, col0, acc[rt], nullptr, nullptr, 0, 0);
  __syncthreads();
  // h1 = silu([h_v | m_v] @ nu_w0^T + b0) -> sB   (K=256 split in two segments)
  zero4(acc);
  seg128(acc, hv_b, 128, nu_w0, 256, col0, 0);
  seg128(acc, sA, 128, nu_w0, 256, col0, 128);
  for (int rt = 0; rt < 4; ++rt)
    store_tile(sB, 128, rt * 16, col0, acc[rt], nu_b0, nullptr, 0, 1);
  __syncthreads();
  // h2 -> sA
  zero4(acc);
  seg128(acc, sB, 128, nu_w1, 128, col0, 0);
  for (int rt = 0; rt < 4; ++rt)
    store_tile(sA, 128, rt * 16, col0, acc[rt], nu_b1, nullptr, 0, 1);
  __syncthreads();
  // h_v_new = h_v + h2 @ nu_w2^T + b2 -> sB
  zero4(acc);
  seg128(acc, sA, 128, nu_w2, 128, col0, 0);
  for (int rt = 0; rt < 4; ++rt)
    store_tile(sB, 128, rt * 16, col0, acc[rt], nu_b2, hv_b, 128, 0);
  __syncthreads();
  for (int idx = tid; idx < 8192; idx += 256)
    hvn[(size_t)b * 8192 + idx] = sB[idx];
  if (tid < 128) {  // dx_raw = tanh(dx_w @ h_v_new + dx_b), 64 rows x 2 cols
    const int row = tid >> 1;
    const int c = tid & 1;
    float a = dx_b[c];
    const float* wr = dx_w + c * 128;
    for (int k = 0; k < 128; ++k) a += wr[k] * sB[row * 128 + k];
    dxr[((size_t)b * 64 + row) * 2 + c] = tanhf(a);
  }
}

__device__ float block_reduce_256(float v, float* s_red) {
  const int tid = threadIdx.x;
  s_red[tid] = v;
  __syncthreads();
  for (int off = 128; off > 0; off >>= 1) {
    if (tid < off) s_red[tid] += s_red[tid + off];
    __syncthreads();
  }
  const float r = s_red[0];
  __syncthreads();
  return r;
}

// -------- Kernel 4: per-batch finalize (dx, Jastrow MLP, envelope, cusps) -----
__global__ __launch_bounds__(256) void k4_final(const float* __restrict__ x,
                                                const int* __restrict__ spin,
                                                const float* __restrict__ stri,
                                                const float* __restrict__ hvn,
                                                const float* __restrict__ dxr,
                                                const float* bf_scale_raw,
                                                const float* f_w0, const float* f_b0,
                                                const float* f_w1, const float* f_b1,
                                                const float* f_w2, const float* f_b2,
                                                float* __restrict__ out) {
  __shared__ float s_fin[258];
  __shared__ float s_f1[64];
  __shared__ float s_f2[64];
  __shared__ float s_red[256];
  __shared__ float s_dxm[2];
  const int b = blockIdx.x;
  const int tid = threadIdx.x;
  if (tid < 128) {  // h_v_mean
    float s = 0.0f;
    for (int i = 0; i < 64; ++i) s += hvn[((size_t)b * 64 + i) * 128 + tid];
    s_fin[tid] = s * (1.0f / 64.0f);
  } else {          // h_e_mean over P=2016 upper-tri pairs
    const int c = tid - 128;
    float s = 0.0f;
    for (int i = 0; i < 64; ++i) s += stri[((size_t)b * 64 + i) * 128 + c];
    s_fin[128 + c] = s * (1.0f / 2016.0f);
  }
  float px2 = 0.0f;
  if (tid < 128) {
    const float xv = x[(size_t)b * 128 + tid];
    px2 = xv * xv;
  }
  const float sumx2 = block_reduce_256(px2, s_red);
  // pair terms: s1_mean and analytic cusps
  const float eps2 = 0.08f;  // (0.2/sqrt(0.5))^2
  float ps1 = 0.0f, pc = 0.0f;
  for (int p = tid; p < 2016; p += 256) {
    int i0 = 0, rem = p;
    while (rem >= (63 - i0)) { rem -= (63 - i0); ++i0; }
    const int j0 = i0 + 1 + rem;
    const float d0 = x[((size_t)b * 64 + i0) * 2 + 0] - x[((size_t)b * 64 + j0) * 2 + 0];
    const float d1 = x[((size_t)b * 64 + i0) * 2 + 1] - x[((size_t)b * 64 + j0) * 2 + 1];
    const float d2 = d0 * d0 + d1 * d1;
    ps1 += log1pf(d2 / eps2 + 1.0f);  // log1p((r_pairs/eps)^2), r_pairs^2 = d2+eps^2
    const float r = sqrtf(d2 + 1e-30f);
    const float g = (spin[i0] == spin[j0]) ? (1.0f / 3.0f) : 1.0f;  // 1/(D+1), 1/(D-1)
    pc += g * r * __expf(-r);
  }
  const float s1sum = block_reduce_256(ps1, s_red);
  const float csum = block_reduce_256(pc, s_red);
  if (tid == 0) {
    s_fin[256] = 0.5f * sumx2 * (1.0f / 128.0f);  // r2_mean of x_sc
    s_fin[257] = s1sum * (1.0f / 2016.0f);
  }
  __syncthreads();
  if (tid < 64) {
    float a = f_b0[tid];
    const float* wr = f_w0 + tid * 258;
    for (int k = 0; k < 258; ++k) a += wr[k] * s_fin[k];
    s_f1[tid] = silu_f(a);
  }
  __syncthreads();
  if (tid < 64) {
    float a = f_b1[tid];
    const float* wr = f_w1 + tid * 64;
    for (int k = 0; k < 64; ++k) a += wr[k] * s_f1[k];
    s_f2[tid] = silu_f(a);
  }
  __syncthreads();
  if (tid == 0) {
    float a = f_b2[0];
    for (int k = 0; k < 64; ++k) a += f_w2[k] * s_f2[k];
    // envelope: exp(-(sum x^2 / N) / (ENV_W^2/OMEGA)) = exp(-r2pp/18)
    const float env = __expf(-(sumx2 * (1.0f / 64.0f)) * (1.0f / 18.0f));
    out[8192 + b] = a * env + csum;
  }
  if (tid < 2) {
    float m = 0.0f;
    for (int i = 0; i < 64; ++i) m += dxr[((size_t)b * 64 + i) * 2 + tid];
    s_dxm[tid] = m * (1.0f / 64.0f);
  }
  __syncthreads();
  const float sps = log1pf(__expf(bf_scale_raw[0]));  // softplus
  if (tid < 128) {
    const int row = tid >> 1;
    const int c = tid & 1;
    out[((size_t)b * 64 + row) * 2 + c] =
        (dxr[((size_t)b * 64 + row) * 2 + c] - s_dxm[c]) * sps;
  }
}

extern "C" void kernel_launch(void* const* d_in, const int* in_sizes, int n_in,
                              void* d_out, int out_size, void* d_ws, size_t ws_size,
                              hipStream_t stream) {
  const float* x      = (const float*)d_in[0];
  const int*   spin   = (const int*)d_in[1];
  const float* ne_w   = (const float*)d_in[2];
  const float* ne_b   = (const float*)d_in[3];
  const float* ee_w0  = (const float*)d_in[4];
  const float* ee_b0  = (const float*)d_in[5];
  const float* ee_w1  = (const float*)d_in[6];
  const float* ee_b1  = (const float*)d_in[7];
  const float* rv_w   = (const float*)d_in[8];
  const float* eu_w0  = (const float*)d_in[9];
  const float* eu_b0  = (const float*)d_in[10];
  const float* eu_w1  = (const float*)d_in[11];
  const float* eu_b1  = (const float*)d_in[12];
  const float* re_w   = (const float*)d_in[13];
  const float* nu_w0  = (const float*)d_in[14];
  const float* nu_b0  = (const float*)d_in[15];
  const float* nu_w1  = (const float*)d_in[16];
  const float* nu_b1  = (const float*)d_in[17];
  const float* nu_w2  = (const float*)d_in[18];
  const float* nu_b2  = (const float*)d_in[19];
  const float* dx_w   = (const float*)d_in[20];
  const float* dx_b   = (const float*)d_in[21];
  const float* bf     = (const float*)d_in[22];
  const float* f_w0   = (const float*)d_in[23];
  const float* f_b0   = (const float*)d_in[24];
  const float* f_w1   = (const float*)d_in[25];
  const float* f_b1   = (const float*)d_in[26];
  const float* f_w2   = (const float*)d_in[27];
  const float* f_b2   = (const float*)d_in[28];

  float* ws   = (float*)d_ws;
  float* hv   = ws;                 // 64*64*128
  float* v    = ws + 524288;        // 64*64*128
  float* smv  = ws + 1048576;       // 64*64*128  sum_{j!=i} h_e
  float* stri = ws + 1572864;       // 64*64*128  sum_{j>i}  h_e
  float* hvn  = ws + 2097152;       // 64*64*128  updated node features
  float* dxr  = ws + 2621440;       // 64*64*2    raw tanh backflow
  float* out  = (float*)d_out;      // 8192 dx + 64 f

  k1_node<<<64, 256, 0, stream>>>(x, spin, ne_w, ne_b, rv_w, hv, v);
  k2_edge<<<4096, 256, 0, stream>>>(x, v, ee_w0, ee_b0, ee_w1, ee_b1,
                                    eu_w0, eu_b0, eu_w1, eu_b1, smv, stri);
  k3_nodeupd<<<64, 256, 0, stream>>>(hv, smv, re_w, nu_w0, nu_b0, nu_w1, nu_b1,
                                     nu_w2, nu_b2, dx_w, dx_b, hvn, dxr);
  k4_final<<<64, 256, 0, stream>>>(x, spin, stri, hvn, dxr, bf,
                                   f_w0, f_b0, f_w1, f_b1, f_w2, f_b2, out);
}